// MobileMQA_87995289960577
// MI455X (gfx1250) — compile-verified
//
#include <hip/hip_runtime.h>
#include <hip/hip_bf16.h>

typedef __attribute__((ext_vector_type(2))) float v2f;
typedef __attribute__((ext_vector_type(8))) float v8f;

#define BATCH 32
#define CDIM  512
#define NTOK  1024
#define HD    64

// Workspace layout (bytes):
//   [0, 131072)            wvT : 64 x 512 f32, wvT[d*512 + c]  (= W_v_eff transposed)
//   [131072, 131328)       bv  : 64 f32 (head-averaged V bias)
//   [131328, 131328+8MB)   YT  : per batch 512x128 f32, YT[b][k*128 + r]
//                                (k = s*64+d scrambled col, r = unique proj row)

// ---------------------------------------------------------------------------
// Kernel 1: fold the head-mean into the V weights (mean is linear).
// ---------------------------------------------------------------------------
__global__ void mqa_prep_kernel(const float* __restrict__ Wqkv,
                                const float* __restrict__ bqkv,
                                float* __restrict__ wvT,
                                float* __restrict__ bv) {
    int i = blockIdx.x * blockDim.x + threadIdx.x;   // 0 .. 32767
    if (i < CDIM * HD) {
        int c = i >> 6;          // 0..511  (consecutive threads -> consecutive d: coalesced loads)
        int d = i & (HD - 1);    // 0..63
        float s = 0.f;
        #pragma unroll
        for (int h = 0; h < 8; ++h)
            s += Wqkv[c * (3 * CDIM) + 2 * CDIM + h * HD + d];
        wvT[d * CDIM + c] = 0.125f * s;
    }
    if (i < HD) {
        float s = 0.f;
        #pragma unroll
        for (int h = 0; h < 8; ++h)
            s += bqkv[2 * CDIM + h * HD + i];
        bv[i] = 0.125f * s;
    }
}

// ---------------------------------------------------------------------------
// Kernel 2: v_mean^T[b] (64x1024) = W_v_eff^T (64x512) @ x_b (512x1024)
// f32 WMMA 16x16x4. A panel (16 d-rows x 512) staged in LDS (padded).
// Epilogue writes directly into the scrambled-proj layout YT[b][ (s*64+d)*128 + r ]
// where the token n = r*8 + s, with the V bias added.
// ---------------------------------------------------------------------------
__global__ void mqa_vmean_kernel(const float* __restrict__ x,
                                 const float* __restrict__ wvT,
                                 const float* __restrict__ bv,
                                 float* __restrict__ YT) {
    __shared__ float lA[16 * 513];           // 513-float row stride: conflict-free ds reads

    int blk = blockIdx.x;                    // 1024 blocks
    int b   = blk >> 5;                      // 32 blocks per batch
    int rem = blk & 31;
    int mt  = rem >> 3;                      // which 16-row quarter of the 64 d-dims
    int ng  = rem & 7;                       // n-tile group (8 waves each)
    int tid = threadIdx.x;

    // Stage the A panel: rows mt*16 .. mt*16+15 of wvT, all 512 k.
    for (int i = tid; i < 16 * 512; i += 256) {
        int row = i >> 9;
        int c   = i & 511;
        lA[row * 513 + c] = wvT[(mt * 16 + row) * CDIM + c];
    }
    __syncthreads();

    int wave = tid >> 5;
    int lane = tid & 31;
    int nl   = lane & 15;                    // A: M row / B: N col within tile
    int kh   = (lane >> 4) << 1;             // lane-half selects K pair {0,1} vs {2,3}
    int n    = (ng * 8 + wave) * 16 + nl;    // token index 0..1023
    const float* __restrict__ xb = x + (size_t)b * CDIM * NTOK;

    v8f acc = {};
    for (int k = 0; k < CDIM; k += 4) {
        v2f a, bb;
        a.x  = lA[nl * 513 + k + kh];
        a.y  = lA[nl * 513 + k + kh + 1];
        bb.x = xb[(size_t)(k + kh) * NTOK + n];
        bb.y = xb[(size_t)(k + kh + 1) * NTOK + n];
        __builtin_prefetch(&xb[(size_t)(k + kh + 8) * NTOK + n], 0, 1);
        acc = __builtin_amdgcn_wmma_f32_16x16x4_f32(
                  false, a, false, bb, (short)0, acc, false, false);
    }

    // D layout: acc[r] holds M = r + 8*(lane>=16), N = lane&15.
    float* __restrict__ yb = YT + (size_t)b * (CDIM * 128);
    int r = n >> 3;                          // unique proj row 0..127
    int s = n & 7;                           // scramble slot 0..7
    int mhi = (lane >> 4) << 3;
    #pragma unroll
    for (int racc = 0; racc < 8; ++racc) {
        int d = mt * 16 + racc + mhi;        // d-dim 0..63
        float val = acc[racc] + bv[d];
        yb[(s * HD + d) * 128 + r] = val;
    }
}

// ---------------------------------------------------------------------------
// Kernel 3: Z^T[b] (512x128) = W_proj^T (512x512) @ Y^T (512x128), + b_proj.
// f32 WMMA 16x16x4. Output rows repeat with period 128 in the reference
// (Z[b,m] == Z[b, m mod 128]), so each result is written to 8 final positions:
// out[b, c, rep*128 + r] — lanes cover consecutive r => 64B-coalesced stores.
// ---------------------------------------------------------------------------
__global__ void mqa_proj_kernel(const float* __restrict__ Wp,
                                const float* __restrict__ bp,
                                const float* __restrict__ YT,
                                float* __restrict__ out) {
    int blk  = blockIdx.x;                   // 1024 blocks
    int b    = blk >> 5;
    int rem  = blk & 31;
    int tid  = threadIdx.x;
    int wave = tid >> 5;
    int lane = tid & 31;
    int tile = rem * 8 + wave;               // 0..255 tiles per batch
    int c0   = (tile >> 3) << 4;             // 32 c-tiles
    int r0   = (tile & 7) << 4;              // 8 r-tiles
    int nl   = lane & 15;
    int kh   = (lane >> 4) << 1;
    const float* __restrict__ yb = YT + (size_t)b * (CDIM * 128);

    v8f acc = {};
    for (int k = 0; k < CDIM; k += 4) {
        v2f a, bb;
        // A[m=c][k] = W_proj[k][c]  (implicit transpose via indexing)
        a.x  = Wp[(size_t)(k + kh) * CDIM + c0 + nl];
        a.y  = Wp[(size_t)(k + kh + 1) * CDIM + c0 + nl];
        // B[k][n=r] = Y^T[k][r]  — lanes read consecutive r: coalesced
        bb.x = yb[(k + kh) * 128 + r0 + nl];
        bb.y = yb[(k + kh + 1) * 128 + r0 + nl];
        acc = __builtin_amdgcn_wmma_f32_16x16x4_f32(
                  false, a, false, bb, (short)0, acc, false, false);
    }

    float* __restrict__ ob = out + (size_t)b * CDIM * NTOK;
    int mhi = (lane >> 4) << 3;
    #pragma unroll
    for (int racc = 0; racc < 8; ++racc) {
        int c = c0 + racc + mhi;
        float val = acc[racc] + bp[c];
        size_t base = (size_t)c * NTOK + (size_t)(r0 + nl);
        #pragma unroll
        for (int rep = 0; rep < 8; ++rep)
            ob[base + (size_t)rep * 128] = val;
    }
}

// ---------------------------------------------------------------------------
extern "C" void kernel_launch(void* const* d_in, const int* in_sizes, int n_in,
                              void* d_out, int out_size, void* d_ws, size_t ws_size,
                              hipStream_t stream) {
    const float* x    = (const float*)d_in[0];
    const float* Wqkv = (const float*)d_in[1];
    const float* bqkv = (const float*)d_in[2];
    const float* Wp   = (const float*)d_in[3];
    const float* bp   = (const float*)d_in[4];
    float* out = (float*)d_out;

    char*  ws  = (char*)d_ws;
    float* wvT = (float*)(ws);
    float* bv  = (float*)(ws + 131072);
    float* YT  = (float*)(ws + 131328);

    mqa_prep_kernel<<<128, 256, 0, stream>>>(Wqkv, bqkv, wvT, bv);
    mqa_vmean_kernel<<<1024, 256, 0, stream>>>(x, wvT, bv, YT);
    mqa_proj_kernel<<<1024, 256, 0, stream>>>(Wp, bp, YT, out);
}